// STGCNN_35338990911692
// MI455X (gfx1250) — compile-verified
//
#include <hip/hip_runtime.h>

typedef float v2f __attribute__((ext_vector_type(2)));
typedef float v8f __attribute__((ext_vector_type(8)));

#define BP_   128        // B*P
#define N_    2000
#define E_    32000
#define C_    128        // C_IN == C_OUT
#define CH_   32         // channel chunk held in LDS by the scatter kernel
#define SUB_  (CH_ / 4)  // float4 sub-slots per edge (== 8)

// ---------------------------------------------------------------------------
// Kernel 1: per-(b,p) degree histogram (with self loop) -> dinv = rsqrt(deg)
// ---------------------------------------------------------------------------
__global__ void gcn_deg_kernel(const int* __restrict__ edges,
                               float* __restrict__ dinv)
{
    __shared__ float sdeg[N_];                       // 8 KB
    const int bp  = blockIdx.x;
    const int tid = threadIdx.x;
    const int nt  = blockDim.x;

    for (int i = tid; i < N_; i += nt) sdeg[i] = 1.0f;   // self-loop
    __syncthreads();

    const int* ei = edges + (size_t)bp * E_ * 2;
    for (int e = tid; e < E_; e += nt) {
        const int dst = ei[2 * e + 1];
        atomicAdd(&sdeg[dst], 1.0f);                 // ds_add_f32
    }
    __syncthreads();

    float* dv = dinv + (size_t)bp * N_;
    for (int i = tid; i < N_; i += nt) dv[i] = rsqrtf(sdeg[i]);
}

// ---------------------------------------------------------------------------
// Kernel 2: XW = X @ W as a flat [256000,128] x [128,128] fp32 WMMA GEMM.
// One 16-row tile per block; wave w (of 8) computes column tile w.
// K swept in steps of 4 with v_wmma_f32_16x16x4_f32 (exact fp32 math).
// ---------------------------------------------------------------------------
__global__ void gcn_gemm_wmma_kernel(const float* __restrict__ X,
                                     const float* __restrict__ W,
                                     float* __restrict__ XW)
{
    const int lane = threadIdx.x & 31;
    const int wave = threadIdx.x >> 5;       // 0..7 -> 16-wide column tile
    const int rt   = blockIdx.x;             // 16-row tile of flat rows
    const int half = lane >> 4;              // 0: K={0,1}, 1: K={2,3}
    const int l16  = lane & 15;

    const int row = rt * 16 + l16;           // A-matrix row this lane supplies
    const int col = wave * 16 + l16;         // B-matrix column this lane supplies

    const float* arow = X + (size_t)row * C_ + half * 2;
    const float* bcol = W + (size_t)(half * 2) * C_ + col;

    v8f acc = {};
    #pragma unroll
    for (int k0 = 0; k0 < C_; k0 += 4) {
        v2f a, b;
        a.x = arow[k0];
        a.y = arow[k0 + 1];
        b.x = bcol[(size_t)k0 * C_];
        b.y = bcol[(size_t)(k0 + 1) * C_];
        // D = A(16x4) * B(4x16) + C   (8 args: neg_a, A, neg_b, B, c_mod, C, reuse_a, reuse_b)
        acc = __builtin_amdgcn_wmma_f32_16x16x4_f32(false, a, false, b,
                                                    (short)0, acc, false, false);
    }

    // C/D layout: VGPR r -> lanes 0-15 hold M=r, lanes 16-31 hold M=8+r, N=lane&15
    const int mbase = rt * 16 + half * 8;
    float* outp = XW + (size_t)mbase * C_ + col;
    #pragma unroll
    for (int r = 0; r < 8; ++r)
        outp[(size_t)r * C_] = acc[r];
}

// ---------------------------------------------------------------------------
// Kernel 3: edge scatter + self-loop + bias + PReLU, per (bp, channel-chunk).
// LDS accumulator [N_ x CH_] (256 KB dynamic), ds_add_f32 atomics for edges.
// xw_out is BOTH the XW input (from kernel 2) and the final output buffer:
// each block reads then overwrites only its own (bp, chunk) slice.
// ---------------------------------------------------------------------------
__global__ void gcn_scatter_kernel(float* __restrict__ xw_out,
                                   const int* __restrict__ edges,
                                   const float* __restrict__ dinv,
                                   const float* __restrict__ bias,
                                   const float* __restrict__ prelu)
{
    extern __shared__ float acc[];           // N_ * CH_ floats
    const int bp  = blockIdx.x;
    const int c0  = blockIdx.y * CH_;
    const int tid = threadIdx.x;
    const int nt  = blockDim.x;

    for (int i = tid; i < N_ * CH_; i += nt) acc[i] = 0.0f;
    __syncthreads();

    const int*   ei = edges  + (size_t)bp * E_ * 2;
    const float* dv = dinv   + (size_t)bp * N_;
    const float* xw = xw_out + (size_t)bp * N_ * C_;

    const int total = E_ * SUB_;             // 8 float4 slots per edge
    for (int idx = tid; idx < total; idx += nt) {
        const int e   = idx >> 3;            // idx / SUB_
        const int sub = idx & (SUB_ - 1);
        __builtin_prefetch(ei + 2 * (e + (nt >> 3)), 0, 0);   // global_prefetch_b8
        const int src = ei[2 * e];
        const int dst = ei[2 * e + 1];
        const float norm = dv[src] * dv[dst];
        const float4 v = *reinterpret_cast<const float4*>(
            xw + (size_t)src * C_ + c0 + sub * 4);
        float* slot = acc + dst * CH_ + sub * 4;
        atomicAdd(slot + 0, norm * v.x);
        atomicAdd(slot + 1, norm * v.y);
        atomicAdd(slot + 2, norm * v.z);
        atomicAdd(slot + 3, norm * v.w);
    }
    __syncthreads();

    const float a = prelu[0];
    float* out = xw_out + (size_t)bp * N_ * C_;
    for (int idx = tid; idx < N_ * SUB_; idx += nt) {
        const int n   = idx >> 3;
        const int sub = idx & (SUB_ - 1);
        const int c   = sub * 4;
        const float dn   = dv[n];
        const float self = dn * dn;          // self-loop norm
        const float4 xv = *reinterpret_cast<const float4*>(
            xw + (size_t)n * C_ + c0 + c);
        float4 r;
        r.x = acc[n * CH_ + c + 0] + self * xv.x + bias[c0 + c + 0];
        r.y = acc[n * CH_ + c + 1] + self * xv.y + bias[c0 + c + 1];
        r.z = acc[n * CH_ + c + 2] + self * xv.z + bias[c0 + c + 2];
        r.w = acc[n * CH_ + c + 3] + self * xv.w + bias[c0 + c + 3];
        r.x = r.x >= 0.0f ? r.x : a * r.x;
        r.y = r.y >= 0.0f ? r.y : a * r.y;
        r.z = r.z >= 0.0f ? r.z : a * r.z;
        r.w = r.w >= 0.0f ? r.w : a * r.w;
        *reinterpret_cast<float4*>(out + (size_t)n * C_ + c0 + c) = r;
    }
}

// ---------------------------------------------------------------------------
extern "C" void kernel_launch(void* const* d_in, const int* in_sizes, int n_in,
                              void* d_out, int out_size, void* d_ws, size_t ws_size,
                              hipStream_t stream) {
    const float* X     = (const float*)d_in[0];   // [B,P,N,C_IN]
    const int*   edges = (const int*)  d_in[1];   // [B,P,E,2]
    const float* W     = (const float*)d_in[2];   // [C_IN,C_OUT]
    const float* bias  = (const float*)d_in[3];   // [C_OUT]
    const float* prelu = (const float*)d_in[4];   // [1]
    float* out = (float*)d_out;                   // [B,P,N,C_OUT] (also XW scratch)
    float* dinv = (float*)d_ws;                   // BP_*N_ floats (~1 MB)

    // 1) degrees -> dinv
    gcn_deg_kernel<<<dim3(BP_), 256, 0, stream>>>(edges, dinv);

    // 2) XW via fp32 WMMA, written straight into d_out
    gcn_gemm_wmma_kernel<<<dim3((BP_ * N_) / 16), 256, 0, stream>>>(X, W, out);

    // 3) scatter + self-loop + bias + PReLU, in place over d_out
    gcn_scatter_kernel<<<dim3(BP_, C_ / CH_), 256,
                         (size_t)N_ * CH_ * sizeof(float), stream>>>(
        out, edges, dinv, bias, prelu);
}